// ULTIMUS_6485400617397
// MI455X (gfx1250) — compile-verified
//
#include <hip/hip_runtime.h>

typedef float v2f __attribute__((ext_vector_type(2)));
typedef float v8f __attribute__((ext_vector_type(8)));

#define TPB 256
#define NWAVES_WG (TPB / 32)
// padded LDS row strides (floats) chosen for conflict-free strided access mod 64 banks
#define XS 52
#define KSTR 36
#define ZS 12

__device__ __forceinline__ v8f wmma4(v2f a, v2f b, v8f c) {
  // (neg_a, A, neg_b, B, c_mod, C, reuse_a, reuse_b)
  return __builtin_amdgcn_wmma_f32_16x16x4_f32(false, a, false, b, (short)0, c,
                                               false, false);
}

__global__ __launch_bounds__(TPB) void ultimus_attn_kernel(
    const float* __restrict__ x,
    const float* __restrict__ Wk, const float* __restrict__ bk,
    const float* __restrict__ Wq, const float* __restrict__ bq,
    const float* __restrict__ Wv, const float* __restrict__ bv,
    const float* __restrict__ Wz, const float* __restrict__ bz,
    float* __restrict__ out, int npairs) {
  __shared__ __align__(16) float lx[NWAVES_WG * 16 * XS];
  __shared__ __align__(16) float lk[NWAVES_WG * 16 * KSTR];
  __shared__ __align__(16) float lz[NWAVES_WG * 16 * ZS];

  const int lane = threadIdx.x & 31;
  const int wave = threadIdx.x >> 5;
  float* xb = lx + wave * 16 * XS;
  float* kqvb = lk + wave * 16 * KSTR;
  float* zb = lz + wave * 16 * ZS;

  const int hi = lane >> 4;  // 0: rows/k-low half, 1: rows+8/k-high half
  const int lo = lane & 15;  // tile row (A) or tile column (B/C/D)
  const int d = lane & 7;

  // ---- loop-invariant WMMA B fragments (weights), column-per-lane layout ----
  // Phase-1 B = Wc^T, Wc = [Wk; Wq; Wv] (24x48). Tile0 cols = K|Q, tile1 = V|0.
  v2f b1a[12], b1b[12];
  {
    const float* rKQ = (lo < 8) ? (Wk + lo * 48) : (Wq + (lo - 8) * 48);
    const float* rV = Wv + (lo & 7) * 48;
    const float vmask = (lo < 8) ? 1.f : 0.f;
#pragma unroll
    for (int ks = 0; ks < 12; ++ks) {
      const int k0 = ks * 4 + 2 * hi;
      v2f t0 = {rKQ[k0], rKQ[k0 + 1]};
      v2f t1 = {vmask * rV[k0], vmask * rV[k0 + 1]};
      b1a[ks] = t0;
      b1b[ks] = t1;
    }
  }
  // Phase-3 B = Wz^T (8x48): B[k,n] = Wz[n,k]
  v2f b3[3][2];
#pragma unroll
  for (int t = 0; t < 3; ++t)
#pragma unroll
    for (int ks = 0; ks < 2; ++ks) {
      const int n = t * 16 + lo, k0 = ks * 4 + 2 * hi;
      v2f tv = {Wz[n * 8 + k0], Wz[n * 8 + k0 + 1]};
      b3[t][ks] = tv;
    }
  const float biasKQ = (lo < 8) ? bk[lo] : bq[lo - 8];
  const float biasV = (lo < 8) ? bv[lo] : 0.f;
  const float biasZ0 = bz[lo], biasZ1 = bz[16 + lo], biasZ2 = bz[32 + lo];
  const float SCALE = 0.3535533905932738f;  // 1/sqrt(8)

  const int nwaves = (gridDim.x * blockDim.x) >> 5;
  const int gw = (blockIdx.x * blockDim.x + threadIdx.x) >> 5;

  for (int p = gw; p < npairs; p += nwaves) {
    const float4* xp = (const float4*)(x + (size_t)p * 768);
    if (p + nwaves < npairs)
      __builtin_prefetch((const char*)(x + (size_t)(p + nwaves) * 768) + lane * 96, 0, 3);

    // ---- stage x tile (16x48 f32) into LDS, coalesced b128 ----
#pragma unroll
    for (int i = 0; i < 6; ++i) {
      const int idx = i * 32 + lane;      // float4 index, 12 per row
      const int r = idx / 12, c4 = idx % 12;
      float4 v = xp[idx];
      *(float4*)&xb[r * XS + c4 * 4] = v;
    }
    asm volatile("s_wait_dscnt 0x0" ::: "memory");

    // ---- phase 1: KQV = X @ Wc^T  (24 x v_wmma_f32_16x16x4_f32) ----
    v8f c0 = {0, 0, 0, 0, 0, 0, 0, 0};
    v8f c1 = {0, 0, 0, 0, 0, 0, 0, 0};
#pragma unroll
    for (int ks = 0; ks < 12; ++ks) {
      v2f a = *(const v2f*)&xb[lo * XS + ks * 4 + 2 * hi];
      c0 = wmma4(a, b1a[ks], c0);
      c1 = wmma4(a, b1b[ks], c1);
    }
#pragma unroll
    for (int r = 0; r < 8; ++r) {
      const int row = r + 8 * hi;
      kqvb[row * KSTR + lo] = c0[r] + biasKQ;        // cols 0-7 K, 8-15 Q
      kqvb[row * KSTR + 16 + lo] = c1[r] + biasV;    // cols 16-23 V (24-31 pad)
    }
    asm volatile("s_wait_dscnt 0x0" ::: "memory");

    // ---- phase 2 (VALU): scores = Q^T K * scale ; Z = scores @ V ----
    {
      const int m0 = hi * 8;
      float4 sA = {0, 0, 0, 0}, sB = {0, 0, 0, 0};
#pragma unroll
      for (int s = 0; s < 8; ++s) {
        const float q = kqvb[(m0 + s) * KSTR + 8 + d];
        float4 kA = *(const float4*)&kqvb[(m0 + s) * KSTR + 0];
        float4 kB = *(const float4*)&kqvb[(m0 + s) * KSTR + 4];
        sA.x += q * kA.x; sA.y += q * kA.y; sA.z += q * kA.z; sA.w += q * kA.w;
        sB.x += q * kB.x; sB.y += q * kB.y; sB.z += q * kB.z; sB.w += q * kB.w;
      }
      float sc[8] = {sA.x * SCALE, sA.y * SCALE, sA.z * SCALE, sA.w * SCALE,
                     sB.x * SCALE, sB.y * SCALE, sB.z * SCALE, sB.w * SCALE};
      float4 zA = {0, 0, 0, 0}, zB = {0, 0, 0, 0};
#pragma unroll
      for (int e = 0; e < 8; ++e) {
        const float se = sc[e];
        float4 vA = *(const float4*)&kqvb[(m0 + e) * KSTR + 16];
        float4 vB = *(const float4*)&kqvb[(m0 + e) * KSTR + 20];
        zA.x += se * vA.x; zA.y += se * vA.y; zA.z += se * vA.z; zA.w += se * vA.w;
        zB.x += se * vB.x; zB.y += se * vB.y; zB.z += se * vB.z; zB.w += se * vB.w;
      }
      // lanes l and l+8 are duplicates writing identical data: benign
      *(float4*)&zb[(m0 + d) * ZS + 0] = zA;
      *(float4*)&zb[(m0 + d) * ZS + 4] = zB;
    }
    asm volatile("s_wait_dscnt 0x0" ::: "memory");

    // ---- phase 3: out = Z @ Wz^T + bz  (6 x v_wmma_f32_16x16x4_f32) ----
    v8f o0 = {0, 0, 0, 0, 0, 0, 0, 0};
    v8f o1 = {0, 0, 0, 0, 0, 0, 0, 0};
    v8f o2 = {0, 0, 0, 0, 0, 0, 0, 0};
#pragma unroll
    for (int ks = 0; ks < 2; ++ks) {
      v2f a = *(const v2f*)&zb[lo * ZS + ks * 4 + 2 * hi];
      o0 = wmma4(a, b3[0][ks], o0);
      o1 = wmma4(a, b3[1][ks], o1);
      o2 = wmma4(a, b3[2][ks], o2);
    }
    float* op = out + (size_t)p * 768;
#pragma unroll
    for (int r = 0; r < 8; ++r) {
      const int row = r + 8 * hi;
      op[row * 48 + lo] = o0[r] + biasZ0;
      op[row * 48 + 16 + lo] = o1[r] + biasZ1;
      op[row * 48 + 32 + lo] = o2[r] + biasZ2;
    }
  }
}

extern "C" void kernel_launch(void* const* d_in, const int* in_sizes, int n_in,
                              void* d_out, int out_size, void* d_ws, size_t ws_size,
                              hipStream_t stream) {
  (void)n_in; (void)out_size; (void)d_ws; (void)ws_size;
  const float* x = (const float*)d_in[0];
  const float* Wk = (const float*)d_in[1];
  const float* bk = (const float*)d_in[2];
  const float* Wq = (const float*)d_in[3];
  const float* bq = (const float*)d_in[4];
  const float* Wv = (const float*)d_in[5];
  const float* bv = (const float*)d_in[6];
  const float* Wz = (const float*)d_in[7];
  const float* bz = (const float*)d_in[8];
  float* out = (float*)d_out;

  const int B = in_sizes[0] / (8 * 48);  // 131072
  const int npairs = B / 2;              // 65536
  ultimus_attn_kernel<<<1024, TPB, 0, stream>>>(x, Wk, bk, Wq, bq, Wv, bv, Wz,
                                                bz, out, npairs);
}